// MiniBatchKmeans_14156212208089
// MI455X (gfx1250) — compile-verified
//
#include <hip/hip_runtime.h>
#include <hip/hip_bf16.h>

// ---------------------------------------------------------------------------
// MiniBatch K-means, one iteration.  N=500000, D=64, K=512, fp32.
// Distance GEMM on V_WMMA_F32_16X16X4_F32 (CDNA5 fp32 matrix path).
// Each wave owns TWO 16-row M-tiles so every B operand fetched from LDS
// feeds two WMMAs (halves LDS bandwidth per FLOP, interleaves two
// accumulator chains for XDL pipelining).
// ---------------------------------------------------------------------------

typedef __attribute__((ext_vector_type(2))) float v2f;
typedef __attribute__((ext_vector_type(8))) float v8f;

constexpr int Dk = 64;    // feature dim
constexpr int Kc = 512;   // number of centroids
constexpr int WAVES_PER_BLOCK = 8;     // 256 threads, 8 wave32s
constexpr int ROWS_PER_WAVE   = 32;    // two WMMA M-tiles per wave

// ---------------------------------------------------------------------------
// Kernel 1: gather centroids C = X[init_idx], compute c2[k] = ||c_k||^2,
// zero sums[K*D] and counts[K].  grid = K blocks of 64 threads.
// ---------------------------------------------------------------------------
__global__ void kmeans_prep(const float* __restrict__ X,
                            const int* __restrict__ init_idx,
                            float* __restrict__ C,
                            float* __restrict__ c2,
                            float* __restrict__ sums,
                            float* __restrict__ counts) {
    const int k = blockIdx.x;
    const int t = threadIdx.x;           // 0..63
    const int src = init_idx[k];
    const float v = X[(size_t)src * Dk + t];
    C[k * Dk + t]    = v;
    sums[k * Dk + t] = 0.0f;

    __shared__ float red[Dk];
    red[t] = v * v;
    __syncthreads();
    if (t < 32) {
        float s = red[t] + red[t + 32];
        #pragma unroll
        for (int off = 16; off > 0; off >>= 1)
            s += __shfl_down(s, off, 32);
        if (t == 0) {
            c2[k]     = s;
            counts[k] = 0.0f;
        }
    }
}

// ---------------------------------------------------------------------------
// Kernel 2: per-wave 32-row tile of X, all 512 centroids in LDS.
// Per 16-centroid column tile: preload 16 B pairs (ds_load), then 32 WMMAs
// (two interleaved accumulator chains).  d = c2[n] - 2*dot feeds a running
// argmin; labels reduced via __shfl_xor; rows accumulated with f32 atomics.
// ---------------------------------------------------------------------------
__global__ void __launch_bounds__(256)
kmeans_assign(const float* __restrict__ X,
              const float* __restrict__ C,
              const float* __restrict__ c2,
              float* __restrict__ sums,
              float* __restrict__ counts,
              int numTiles) {
    extern __shared__ float lds[];
    float* ldsC  = lds;              // K*D floats = 128 KB
    float* ldsC2 = lds + Kc * Dk;    // K floats

    // Cooperative load of the centroid matrix into LDS (float4 transfers).
    {
        const int tid = threadIdx.x;                 // 0..255
        const float4* src4 = (const float4*)C;
        float4* dst4 = (float4*)ldsC;
        #pragma unroll 4
        for (int i = tid; i < (Kc * Dk) / 4; i += 256) dst4[i] = src4[i];
        for (int i = tid; i < Kc; i += 256) ldsC2[i] = c2[i];
    }
    __syncthreads();

    const int wave = threadIdx.x >> 5;
    const int lane = threadIdx.x & 31;
    const int tile = blockIdx.x * WAVES_PER_BLOCK + wave;
    if (tile >= numTiles) return;     // wave-uniform: EXEC stays all-1s inside

    const int row0 = tile * ROWS_PER_WAVE;          // first of 32 rows
    const int m  = lane & 15;         // A row / B col / D col within a tile
    const int kb = (lane >> 4) * 2;   // K-pair selector (lanes 16-31 -> K=2,3)

    // --- A-matrices: two 16x64 row tiles, 16 v2f each, reused 32x ---------
    v2f a0[16], a1[16];
    {
        const float* xr0 = X + (size_t)(row0 + m) * Dk + kb;
        const float* xr1 = X + (size_t)(row0 + 16 + m) * Dk + kb;
        #pragma unroll
        for (int j = 0; j < 16; ++j) {
            a0[j] = *(const v2f*)(xr0 + 4 * j);      // global_load_b64
            a1[j] = *(const v2f*)(xr1 + 4 * j);
        }
    }

    float bestVal0[8], bestVal1[8];
    int   bestIdx0[8], bestIdx1[8];
    #pragma unroll
    for (int r = 0; r < 8; ++r) {
        bestVal0[r] = __builtin_inff(); bestIdx0[r] = 0;
        bestVal1[r] = __builtin_inff(); bestIdx1[r] = 0;
    }

    // --- Loop over 32 column tiles of 16 centroids ------------------------
    for (int n0 = 0; n0 < Kc; n0 += 16) {
        // Preload all 16 B pairs for this column tile (batched ds loads).
        v2f b[16];
        {
            const float* crow = ldsC + (n0 + m) * Dk + kb;
            #pragma unroll
            for (int j = 0; j < 16; ++j)
                b[j] = *(const v2f*)(crow + 4 * j);  // ds_load_2addr_b64 pairs
        }
        v8f acc0 = {0.f, 0.f, 0.f, 0.f, 0.f, 0.f, 0.f, 0.f};
        v8f acc1 = {0.f, 0.f, 0.f, 0.f, 0.f, 0.f, 0.f, 0.f};
        #pragma unroll
        for (int j = 0; j < 16; ++j) {
            acc0 = __builtin_amdgcn_wmma_f32_16x16x4_f32(
                false, a0[j], false, b[j], (short)0, acc0, false, false);
            acc1 = __builtin_amdgcn_wmma_f32_16x16x4_f32(
                false, a1[j], false, b[j], (short)0, acc1, false, false);
        }
        const float cc  = ldsC2[n0 + m];   // c2 for column n0 + (lane&15)
        const int   col = n0 + m;
        #pragma unroll
        for (int r = 0; r < 8; ++r) {
            const float d0 = cc - 2.0f * acc0[r];    // x^2 term dropped
            const float d1 = cc - 2.0f * acc1[r];
            if (d0 < bestVal0[r]) { bestVal0[r] = d0; bestIdx0[r] = col; }
            if (d1 < bestVal1[r]) { bestVal1[r] = d1; bestIdx1[r] = col; }
        }
    }

    // --- Argmin reduction across the 16 lanes of each half ----------------
    // VGPR slot r: lanes 0-15 hold row r, lanes 16-31 hold row r+8.
    #pragma unroll
    for (int off = 1; off < 16; off <<= 1) {
        #pragma unroll
        for (int r = 0; r < 8; ++r) {
            float ov = __shfl_xor(bestVal0[r], off, 32);
            int   oi = __shfl_xor(bestIdx0[r], off, 32);
            if (ov < bestVal0[r] || (ov == bestVal0[r] && oi < bestIdx0[r])) {
                bestVal0[r] = ov; bestIdx0[r] = oi;
            }
            ov = __shfl_xor(bestVal1[r], off, 32);
            oi = __shfl_xor(bestIdx1[r], off, 32);
            if (ov < bestVal1[r] || (ov == bestVal1[r] && oi < bestIdx1[r])) {
                bestVal1[r] = ov; bestIdx1[r] = oi;
            }
        }
    }

    // --- Accumulate the 32 rows into sums/counts --------------------------
    #pragma unroll
    for (int r = 0; r < 8; ++r) {
        const int lab[4] = {
            __shfl(bestIdx0[r], 0, 32),   // row row0 + r
            __shfl(bestIdx0[r], 16, 32),  // row row0 + r + 8
            __shfl(bestIdx1[r], 0, 32),   // row row0 + 16 + r
            __shfl(bestIdx1[r], 16, 32)   // row row0 + 24 + r
        };
        const int rowOf[4] = { row0 + r, row0 + r + 8, row0 + 16 + r, row0 + 24 + r };
        #pragma unroll
        for (int q = 0; q < 4; ++q) {
            const float2 xv = ((const float2*)(X + (size_t)rowOf[q] * Dk))[lane];
            atomicAdd(&sums[lab[q] * Dk + 2 * lane + 0], xv.x);
            atomicAdd(&sums[lab[q] * Dk + 2 * lane + 1], xv.y);
            if (lane == 0) atomicAdd(&counts[lab[q]], 1.0f);
        }
    }
}

// ---------------------------------------------------------------------------
// Kernel 3: new_centroids = sums / max(counts, 1)
// ---------------------------------------------------------------------------
__global__ void kmeans_finalize(const float* __restrict__ sums,
                                const float* __restrict__ counts,
                                float* __restrict__ out) {
    const int k = blockIdx.x;
    const int t = threadIdx.x;
    const float cnt = fmaxf(counts[k], 1.0f);
    out[k * Dk + t] = sums[k * Dk + t] / cnt;
}

extern "C" void kernel_launch(void* const* d_in, const int* in_sizes, int n_in,
                              void* d_out, int out_size, void* d_ws, size_t ws_size,
                              hipStream_t stream) {
    const float* X        = (const float*)d_in[0];
    const int*   init_idx = (const int*)d_in[1];
    float*       out      = (float*)d_out;

    const int N        = in_sizes[0] / Dk;        // 500000
    const int numTiles = N / ROWS_PER_WAVE;       // 15625 (N divisible by 32)

    // Workspace layout (floats): C[K*D] | c2[K] | sums[K*D] | counts[K]
    float* ws     = (float*)d_ws;
    float* C      = ws;
    float* c2     = C + Kc * Dk;
    float* sums   = c2 + Kc;
    float* counts = sums + Kc * Dk;

    kmeans_prep<<<Kc, Dk, 0, stream>>>(X, init_idx, C, c2, sums, counts);

    const int blocks = (numTiles + WAVES_PER_BLOCK - 1) / WAVES_PER_BLOCK;
    const size_t shmem = (size_t)(Kc * Dk + Kc) * sizeof(float);   // ~130 KB < 320 KB/WG
    kmeans_assign<<<blocks, 32 * WAVES_PER_BLOCK, shmem, stream>>>(
        X, C, c2, sums, counts, numTiles);

    kmeans_finalize<<<Kc, Dk, 0, stream>>>(sums, counts, out);
}